// CRF_20822001451447
// MI455X (gfx1250) — compile-verified
//
#include <hip/hip_runtime.h>
#include <hip/hip_bf16.h>

// CRF Viterbi decode, B=512 T=512 S=64.
// One 64-thread workgroup per batch element. Per-batch state lives in LDS:
//   - fv double buffer (2x64 f32)
//   - emission ring buffer (8x64 f32), filled by async global->LDS copies
//     with prefetch distance 7 (s_wait_asynccnt 6 retires the oldest)
//   - full backpointer table (511x64 bytes) -> backtrace never touches HBM
//   - mask row (512 f32)
// Transition row (next=tid) and the current fv vector are held in VGPRs so the
// 64-wide max-plus inner loop is pure VALU. Argmax = 8x 8-wide scans + 3-level
// tournament (first-max tie semantics, short dependency chain).

#define VIT_S 64
#define VIT_T 512
#define VIT_NEG_INF (-10000.0f)
#define VIT_PFD 7          // async prefetch distance (outstanding loads)
#define VIT_SLOTS 8        // emission ring slots (power of two)

__global__ __launch_bounds__(VIT_S) void crf_viterbi_kernel(
    const float* __restrict__ logits,   // (B, T, S)
    const float* __restrict__ masks,    // (B, T)
    const float* __restrict__ trans,    // (S, S)  trans[next][prev]
    float* __restrict__ out,            // [B pscore][B x (T-1) seq]
    int B) {
  constexpr int S = VIT_S;
  constexpr int T = VIT_T;

  __shared__ __align__(16) float fvbuf[2][S];
  __shared__ __align__(16) float featbuf[VIT_SLOTS][S];  // 2 KB ring
  __shared__ __align__(16) float maskrow[T];
  __shared__ unsigned char bptr[T - 1][S];               // 32704 B
  __shared__ unsigned char tagseq[T];

  const int tid = threadIdx.x;      // = "next" tag owned by this thread
  const int b = blockIdx.x;
  if (b >= B) return;

  // ---- transition row for next=tid into VGPRs ----
  float trow[S];
  {
    const float4* tr4 = reinterpret_cast<const float4*>(trans + (size_t)tid * S);
#pragma unroll
    for (int i = 0; i < S / 4; ++i) {
      float4 v = tr4[i];
      trow[4 * i + 0] = v.x; trow[4 * i + 1] = v.y;
      trow[4 * i + 2] = v.z; trow[4 * i + 3] = v.w;
    }
  }

  // ---- mask row into LDS ----
  for (int i = tid; i < T; i += S) maskrow[i] = masks[(size_t)b * T + i];

  // ---- init fv: NEG_INF except START_IDX(=0) ----
  fvbuf[0][tid] = (tid == 0) ? 0.0f : VIT_NEG_INF;

  // ---- prime the async emission pipeline: rows t=1..VIT_PFD ----
  const unsigned long long gbase =
      (unsigned long long)(const void*)(logits + (size_t)b * T * S);
  const unsigned int lds_feat0 = (unsigned int)(size_t)(void*)&featbuf[0][tid];
#pragma unroll
  for (int tt = 1; tt <= VIT_PFD; ++tt) {
    unsigned int ldsaddr = lds_feat0 + (unsigned)((tt & (VIT_SLOTS - 1)) * S * 4);
    unsigned int goff = (unsigned int)(((unsigned)tt * S + (unsigned)tid) * 4u);
    asm volatile("global_load_async_to_lds_b32 %0, %1, %2"
                 :: "v"(ldsaddr), "v"(goff), "s"(gbase) : "memory");
  }

  float pscore = VIT_NEG_INF;

  for (int t = 1; t < T; ++t) {
    __syncthreads();                      // fv(t-1) writes visible

    // Retire the oldest in-flight async load: emissions for step t are in
    // featbuf[t & 7] (ASYNCcnt completes in order; <=6 outstanding => oldest
    // of the 7 primed loads has landed).
    asm volatile("s_wait_asynccnt 0x6" ::: "memory");

    const int rd = (t + 1) & 1;           // fv buffer written at step t-1
    const int wr = t & 1;

    // current fv vector into VGPRs
    float fvr[S];
    {
      const float4* fv4 = reinterpret_cast<const float4*>(&fvbuf[rd][0]);
#pragma unroll
      for (int i = 0; i < S / 4; ++i) {
        float4 v = fv4[i];
        fvr[4 * i + 0] = v.x; fvr[4 * i + 1] = v.y;
        fvr[4 * i + 2] = v.z; fvr[4 * i + 3] = v.w;
      }
    }

    const float feat = featbuf[t & (VIT_SLOTS - 1)][tid] * maskrow[t];

    // Issue prefetch for step t+7 into slot (t+7)&7 (== slot consumed at step
    // t-1, retired 8 steps of in-order DS traffic ago). Tail iterations clamp
    // to the last row so exactly 7 loads stay in flight (branch-free wait).
    {
      unsigned int tgt = (unsigned)((t + VIT_PFD < T) ? (t + VIT_PFD) : (T - 1));
      unsigned int ldsaddr =
          lds_feat0 + (unsigned)(((t + VIT_PFD) & (VIT_SLOTS - 1)) * S * 4);
      unsigned int goff = (tgt * (unsigned)S + (unsigned)tid) * 4u;
      asm volatile("global_load_async_to_lds_b32 %0, %1, %2"
                   :: "v"(ldsaddr), "v"(goff), "s"(gbase) : "memory");
    }

    // ---- max-plus over prev: 8 chunked scans + pairwise tournament ----
    // (ties -> lowest index, matching jnp.argmax)
    float cm[8];
    int ci[8];
#pragma unroll
    for (int c = 0; c < 8; ++c) {
      float m0 = fvr[8 * c] + trow[8 * c];
      int i0 = 8 * c;
#pragma unroll
      for (int j = 1; j < 8; ++j) {
        int p = 8 * c + j;
        float s = fvr[p] + trow[p];
        bool gt = s > m0;                 // strict > keeps earliest index
        m0 = gt ? s : m0;
        i0 = gt ? p : i0;
      }
      cm[c] = m0; ci[c] = i0;
    }
#pragma unroll
    for (int off = 1; off < 8; off <<= 1) {
#pragma unroll
      for (int c = 0; c < 8; c += 2 * off) {
        bool gt = cm[c + off] > cm[c];    // left (lower index) wins ties
        cm[c] = gt ? cm[c + off] : cm[c];
        ci[c] = gt ? ci[c + off] : ci[c];
      }
    }
    const float m = cm[0];
    const int idx = ci[0];

    bptr[t - 1][tid] = (unsigned char)idx;
    if (t == T - 1) pscore = m;           // vmax BEFORE emission add (quirk)

    fvbuf[wr][tid] = m + feat;
  }

  // path_score = vmaxs[-1][:, S-1]   (reference quirk: column S-1)
  if (tid == S - 1) out[b] = pscore;

  __syncthreads();                        // backpointer table complete

  // ---- backtrace (serial, entirely in LDS) ----
  if (tid == 0) {
    int tag = bptr[T - 2][S - 1];         // t0 seeded from column S-1
    for (int k = T - 2; k >= 0; --k) {    // re-applies last table once (quirk)
      tagseq[k] = (unsigned char)tag;
      tag = bptr[k][tag];
    }
  }
  __syncthreads();

  // ---- coalesced writeout of best_seq as floats ----
  float* seq_out = out + B + (size_t)b * (T - 1);
  for (int k = tid; k < T - 1; k += S) seq_out[k] = (float)tagseq[k];
}

extern "C" void kernel_launch(void* const* d_in, const int* in_sizes, int n_in,
                              void* d_out, int out_size, void* d_ws, size_t ws_size,
                              hipStream_t stream) {
  const float* logits = (const float*)d_in[0];   // (B, T, S) f32
  const float* masks  = (const float*)d_in[1];   // (B, T)    f32
  const float* trans  = (const float*)d_in[2];   // (S, S)    f32
  float* out = (float*)d_out;

  const int T = VIT_T;
  const int B = in_sizes[1] / T;                 // masks is B*T elements

  crf_viterbi_kernel<<<dim3(B), dim3(VIT_S), 0, stream>>>(logits, masks, trans,
                                                          out, B);
}